// GAT_RL_4810363372828
// MI455X (gfx1250) — compile-verified
//
#include <hip/hip_runtime.h>
#include <hip/hip_bf16.h>

typedef __attribute__((ext_vector_type(2))) float v2f;
typedef __attribute__((ext_vector_type(8))) float v8f;

#define BATCH 128
#define NNODE 64
#define XD 128
#define UVD 512
#define ROWS (BATCH * NNODE)   // 8192
#define NEG_SLOPE 0.04f
#define MASK_VAL -9.0e15f
#define XSP 130                // padded LDS stride for X tile
#define UVP 257                // padded LDS stride for U/V rows
#define SSP 65                 // padded LDS stride for score rows

// ---------------------------------------------------------------------------
// Kernel A: node embeddings. One wave (32 lanes) per node row.
// robot row: 9 -> 64 relu -> 128 relu ; human row: 5 -> 64 relu -> 128 relu
// ---------------------------------------------------------------------------
__global__ __launch_bounds__(256) void embed_kernel(
    const float* __restrict__ robot, const float* __restrict__ human,
    const float* __restrict__ wrW1, const float* __restrict__ wrB1,
    const float* __restrict__ wrW2, const float* __restrict__ wrB2,
    const float* __restrict__ whW1, const float* __restrict__ whB1,
    const float* __restrict__ whW2, const float* __restrict__ whB2,
    float* __restrict__ X)
{
    __shared__ float hid[8][64];
    const int t    = threadIdx.x;
    const int wave = t >> 5;
    const int lane = t & 31;
    const int r    = blockIdx.x * 8 + wave;      // node row in [0, 8192)
    const int b    = r >> 6;
    const int n    = r & 63;

    const float *inp, *W1, *b1, *W2, *b2;
    int din;
    if (n == 0) { din = 9; inp = robot + b * 9;
                  W1 = wrW1; b1 = wrB1; W2 = wrW2; b2 = wrB2; }
    else        { din = 5; inp = human + (b * 63 + (n - 1)) * 5;
                  W1 = whW1; b1 = whB1; W2 = whW2; b2 = whB2; }

    float in_reg[9];
    #pragma unroll
    for (int d = 0; d < 9; ++d) in_reg[d] = (d < din) ? inp[d] : 0.f;

    #pragma unroll
    for (int hh = 0; hh < 2; ++hh) {
        const int hn = lane + 32 * hh;
        float acc = b1[hn];
        for (int d = 0; d < din; ++d) acc = fmaf(in_reg[d], W1[d * 64 + hn], acc);
        hid[wave][hn] = fmaxf(acc, 0.f);
    }
    __syncthreads();

    #pragma unroll
    for (int cc = 0; cc < 4; ++cc) {
        const int c = lane + 32 * cc;
        float acc = b2[c];
        #pragma unroll 8
        for (int k = 0; k < 64; ++k) acc = fmaf(hid[wave][k], W2[k * 128 + c], acc);
        X[r * 128 + c] = fmaxf(acc, 0.f);
    }
}

// ---------------------------------------------------------------------------
// Kernel B: UV = Xin[8192,128] @ Wcat[128,512] (+b1 folded into U columns)
// Wcat cols 0..255   = W1 rows 0..127   (source half)
// Wcat cols 256..511 = W1 rows 128..255 (target half)
// Exact fp32 via V_WMMA_F32_16X16X4_F32: one 16x16 tile per wave, K step 4.
// ---------------------------------------------------------------------------
__global__ __launch_bounds__(256) void uv_gemm_kernel(
    const float* __restrict__ Xin, const float* __restrict__ W1,
    const float* __restrict__ b1, float* __restrict__ UV)
{
    const int t    = threadIdx.x;
    const int wave = t >> 5;
    const int lane = t & 31;
    const int flat = blockIdx.x * 8 + wave;   // tile id 0..16383
    const int tm   = flat >> 5;               // 0..511  (M tiles)
    const int tn   = flat & 31;               // 0..31   (N tiles)
    const int half = lane >> 4;               // 0: K pair 0/1, 1: K pair 2/3
    const int l16  = lane & 15;
    const int mrow = tm * 16 + l16;
    const int ncol = tn * 16 + l16;

    const bool lo = (ncol < 256);
    const float* wb = lo ? (W1 + ncol) : (W1 + 128 * 256 + (ncol - 256));
    const float bias = lo ? b1[ncol & 255] : 0.f;

    const float* arow = Xin + mrow * 128 + 2 * half;

    v8f c = {};
    #pragma unroll 8
    for (int k0 = 0; k0 < 128; k0 += 4) {
        // A frag 16x4: lanes 0-15 -> K=0,1 ; lanes 16-31 -> K=2,3
        v2f a = *(const v2f*)(arow + k0);
        // B frag 4x16: lanes 0-15 -> rows K=0,1 ; lanes 16-31 -> rows K=2,3
        const int kr = k0 + 2 * half;
        v2f bf;
        bf.x = wb[kr * 256];
        bf.y = wb[(kr + 1) * 256];
        c = __builtin_amdgcn_wmma_f32_16x16x4_f32(
                /*neg_a=*/false, a, /*neg_b=*/false, bf,
                /*c_mod=*/(short)0, c, /*reuse_a=*/false, /*reuse_b=*/false);
    }

    // D 16x16 f32: VGPR r -> M=r (lanes 0-15) / M=r+8 (lanes 16-31), N=lane%16
    #pragma unroll
    for (int rr = 0; rr < 8; ++rr) {
        const int row = tm * 16 + rr + 8 * half;
        UV[row * 512 + ncol] = c[rr] + bias;
    }
}

// ---------------------------------------------------------------------------
// Kernel C: GAT attention. Two 256-thread blocks per batch element; each
// block owns 32 attention rows (i), needs full V/X (all j columns).
// score(i,j) = leaky( relu(U[i]+V[j]) . W2 + b2 ), mask (i>=1, j==0),
// softmax over j, then H(rows) = attn @ X via WMMA f32 16x16x4.
// ---------------------------------------------------------------------------
__global__ __launch_bounds__(256) void gat_attn_kernel(
    const float* __restrict__ UV, const float* __restrict__ Xin,
    const float* __restrict__ W2, const float* __restrict__ b2p,
    float* __restrict__ H)
{
    extern __shared__ float smem[];
    float* Xs  = smem;                  // 64*130 = 8320
    float* Us  = Xs + 64 * XSP;         // 32*257 = 8224
    float* Vs  = Us + 32 * UVP;         // 64*257 = 16448
    float* Ss  = Vs + 64 * UVP;         // 32*65  = 2080
    float* W2s = Ss + 32 * SSP;         // 256

    const int t  = threadIdx.x;
    const int b  = blockIdx.x >> 1;
    const int ih = (blockIdx.x & 1) * 32;        // this block's i-row offset
    const float* uvb = UV + b * 64 * 512;

    for (int e = t; e < 64 * 128; e += 256) {
        const int i = e >> 7, c = e & 127;
        Xs[i * XSP + c] = Xin[b * 8192 + e];
    }
    for (int e = t; e < 32 * 256; e += 256) {
        const int i = e >> 8, k = e & 255;
        Us[i * UVP + k] = uvb[(ih + i) * 512 + k];
    }
    for (int e = t; e < 64 * 256; e += 256) {
        const int i = e >> 8, k = e & 255;
        Vs[i * UVP + k] = uvb[i * 512 + 256 + k];
    }
    W2s[t] = W2[t];
    const float b2 = b2p[0];
    __syncthreads();

    // ---- scores: thread owns fixed column j = t&63 and 8 rows i = idx*4+ti.
    // V[j][k] and W2[k] loads hoisted out of the row loop (DS-issue bound
    // otherwise); U reads are wave-uniform broadcasts.
    {
        const int ti = t >> 6;                    // 0..3 (wave-uniform)
        const int j  = t & 63;
        const float* vj = Vs + j * UVP;
        float acc[8] = {};
        for (int k = 0; k < 256; ++k) {
            const float w = W2s[k];
            const float v = vj[k];
            #pragma unroll
            for (int idx = 0; idx < 8; ++idx) {
                const float s = Us[(idx * 4 + ti) * UVP + k] + v;
                acc[idx] = fmaf(fmaxf(s, 0.f), w, acc[idx]);
            }
        }
        #pragma unroll
        for (int idx = 0; idx < 8; ++idx) {
            const int il = idx * 4 + ti;
            const float a = acc[idx] + b2;
            float e = (a > 0.f) ? a : NEG_SLOPE * a;
            if ((ih + il) >= 1 && j == 0) e = MASK_VAL;  // humans !-> robot
            Ss[il * SSP + j] = e;
        }
    }
    __syncthreads();

    // ---- softmax per row (32 rows, one thread each; stride 65 = no conflicts)
    if (t < 32) {
        float* row = Ss + t * SSP;
        float m = row[0];
        for (int j = 1; j < 64; ++j) m = fmaxf(m, row[j]);
        float s = 0.f;
        for (int j = 0; j < 64; ++j) { const float ex = __expf(row[j] - m); row[j] = ex; s += ex; }
        const float inv = 1.f / s;
        for (int j = 0; j < 64; ++j) row[j] *= inv;
    }
    __syncthreads();

    // ---- H(rows ih..ih+31) = attn[32,64] @ X[64,128] via WMMA f32 16x16x4.
    // 16 tiles of 16x16; each of 8 waves does 2 tiles, K stepped by 4.
    {
        const int wave = t >> 5, lane = t & 31;
        const int half = lane >> 4, l16 = lane & 15;
        #pragma unroll
        for (int tt = 0; tt < 2; ++tt) {
            const int tile = wave * 2 + tt;   // 0..15
            const int tm2  = tile >> 3;       // 0..1  (M tile)
            const int tn   = tile & 7;        // 0..7  (N tile)
            const float* arow = Ss + (tm2 * 16 + l16) * SSP + 2 * half;
            const float* bcol = Xs + tn * 16 + l16 + 2 * half * XSP;
            v8f c = {};
            #pragma unroll
            for (int k0 = 0; k0 < 64; k0 += 4) {
                v2f a;  a.x  = arow[k0];          a.y  = arow[k0 + 1];
                v2f bf; bf.x = bcol[k0 * XSP];    bf.y = bcol[k0 * XSP + XSP];
                c = __builtin_amdgcn_wmma_f32_16x16x4_f32(
                        false, a, false, bf, (short)0, c, false, false);
            }
            #pragma unroll
            for (int rr = 0; rr < 8; ++rr) {
                const int row = ih + tm2 * 16 + rr + 8 * half;
                H[b * 8192 + row * 128 + tn * 16 + l16] = c[rr];
            }
        }
    }
}

// ---------------------------------------------------------------------------
// Kernel D: out = X + H1 + H2 (skip connection), float4 vectorized
// ---------------------------------------------------------------------------
__global__ __launch_bounds__(256) void add3_kernel(
    const float4* __restrict__ X, const float4* __restrict__ H1,
    const float4* __restrict__ H2, float4* __restrict__ out)
{
    const int e = blockIdx.x * 256 + threadIdx.x;
    const float4 x = X[e], a = H1[e], bb = H2[e];
    out[e] = make_float4(x.x + a.x + bb.x, x.y + a.y + bb.y,
                         x.z + a.z + bb.z, x.w + a.w + bb.w);
}

// ---------------------------------------------------------------------------
extern "C" void kernel_launch(void* const* d_in, const int* in_sizes, int n_in,
                              void* d_out, int out_size, void* d_ws, size_t ws_size,
                              hipStream_t stream) {
    const float* robot = (const float*)d_in[0];
    const float* human = (const float*)d_in[1];
    const float* wrW1  = (const float*)d_in[2];
    const float* wrB1  = (const float*)d_in[3];
    const float* wrW2  = (const float*)d_in[4];
    const float* wrB2  = (const float*)d_in[5];
    const float* whW1  = (const float*)d_in[6];
    const float* whB1  = (const float*)d_in[7];
    const float* whW2  = (const float*)d_in[8];
    const float* whB2  = (const float*)d_in[9];
    const float* g0W1  = (const float*)d_in[10];
    const float* g0B1  = (const float*)d_in[11];
    const float* g0W2  = (const float*)d_in[12];
    const float* g0B2  = (const float*)d_in[13];
    const float* g1W1  = (const float*)d_in[14];
    const float* g1B1  = (const float*)d_in[15];
    const float* g1W2  = (const float*)d_in[16];
    const float* g1B2  = (const float*)d_in[17];

    float* ws = (float*)d_ws;
    float* X  = ws;                         // 8192*128 = 1048576
    float* H1 = X  + (size_t)ROWS * XD;     // 1048576
    float* H2 = H1 + (size_t)ROWS * XD;     // 1048576
    float* UV = H2 + (size_t)ROWS * XD;     // 8192*512 = 4194304
    float* out = (float*)d_out;

    const size_t attn_lds =
        (size_t)(64 * XSP + 32 * UVP + 64 * UVP + 32 * SSP + 256) * sizeof(float);

    // embeddings -> X
    embed_kernel<<<ROWS / 8, 256, 0, stream>>>(
        robot, human, wrW1, wrB1, wrW2, wrB2, whW1, whB1, whW2, whB2, X);

    // layer 0
    uv_gemm_kernel<<<(ROWS / 16) * (UVD / 16) / 8, 256, 0, stream>>>(X, g0W1, g0B1, UV);
    gat_attn_kernel<<<BATCH * 2, 256, attn_lds, stream>>>(UV, X, g0W2, g0B2, H1);

    // layer 1 (input H1)
    uv_gemm_kernel<<<(ROWS / 16) * (UVD / 16) / 8, 256, 0, stream>>>(H1, g1W1, g1B1, UV);
    gat_attn_kernel<<<BATCH * 2, 256, attn_lds, stream>>>(UV, H1, g1W2, g1B2, H2);

    // skip connection
    add3_kernel<<<(ROWS * XD / 4) / 256, 256, 0, stream>>>(
        (const float4*)X, (const float4*)H1, (const float4*)H2, (float4*)out);
}